// ProbabilisticSurfaceDistanceLoss_15925738734019
// MI455X (gfx1250) — compile-verified
//
#include <hip/hip_runtime.h>
#include <hip/hip_bf16.h>

typedef __attribute__((ext_vector_type(2))) float v2f;
typedef __attribute__((ext_vector_type(8))) float v8f;

// ---------------------------------------------------------------------------
// 1) Face areas: one thread per face, 0.5*|cross(e1,e2)|
// ---------------------------------------------------------------------------
__global__ void psd_face_area_kernel(const float* __restrict__ verts,
                                     const int* __restrict__ faces,
                                     float* __restrict__ areas, int nfaces) {
  int f = blockIdx.x * blockDim.x + threadIdx.x;
  if (f >= nfaces) return;
  int i0 = faces[3 * f + 0], i1 = faces[3 * f + 1], i2 = faces[3 * f + 2];
  float ax = verts[3 * i0 + 0], ay = verts[3 * i0 + 1], az = verts[3 * i0 + 2];
  float bx = verts[3 * i1 + 0], by = verts[3 * i1 + 1], bz = verts[3 * i1 + 2];
  float cx = verts[3 * i2 + 0], cy = verts[3 * i2 + 1], cz = verts[3 * i2 + 2];
  float e1x = bx - ax, e1y = by - ay, e1z = bz - az;
  float e2x = cx - ax, e2y = cy - ay, e2z = cz - az;
  float nx = e1y * e2z - e1z * e2y;
  float ny = e1z * e2x - e1x * e2z;
  float nz = e1x * e2y - e1y * e2x;
  areas[f] = 0.5f * sqrtf(nx * nx + ny * ny + nz * nz);
}

// ---------------------------------------------------------------------------
// 2) In-place inclusive scan (single block, chunked Hillis-Steele + carry)
// ---------------------------------------------------------------------------
__global__ __launch_bounds__(1024) void psd_scan_kernel(float* data, int n) {
  __shared__ float tmp[1024];
  __shared__ float carry_sh;
  int tid = threadIdx.x;
  if (tid == 0) carry_sh = 0.0f;
  __syncthreads();
  for (int base = 0; base < n; base += 1024) {
    int i = base + tid;
    float v = (i < n) ? data[i] : 0.0f;
    tmp[tid] = v;
    __syncthreads();
    for (int off = 1; off < 1024; off <<= 1) {
      float t = (tid >= off) ? tmp[tid - off] : 0.0f;
      __syncthreads();
      tmp[tid] += t;
      __syncthreads();
    }
    float carry = carry_sh;
    if (i < n) data[i] = tmp[tid] + carry;
    __syncthreads();
    if (tid == 1023) carry_sh = carry + tmp[1023];
    __syncthreads();
  }
}

// ---------------------------------------------------------------------------
// 3) Sample points: binary search CDF (searchsorted 'left'), barycentric fold.
//    Output float4 = (x, y, z, |p|^2)
// ---------------------------------------------------------------------------
__global__ void psd_sample_kernel(const float* __restrict__ verts,
                                  const int* __restrict__ faces,
                                  const float* __restrict__ cdf,
                                  const float* __restrict__ u_in,
                                  const float* __restrict__ v_in,
                                  const float* __restrict__ rsel,
                                  int nfaces, int nsamp,
                                  float4* __restrict__ out) {
  int s = blockIdx.x * blockDim.x + threadIdx.x;
  if (s >= nsamp) return;
  float total = cdf[nfaces - 1];
  float target = rsel[s] * total;  // == searchsorted(cdf/total, rsel)
  int lo = 0, hi = nfaces;
  while (lo < hi) {
    int mid = (lo + hi) >> 1;
    if (cdf[mid] < target) lo = mid + 1; else hi = mid;
  }
  int sel = lo < (nfaces - 1) ? lo : (nfaces - 1);

  float u = u_in[s], v = v_in[s];
  float w = 1.0f - u - v;
  if (w < 0.0f) { u = u + w; v = 1.0f - u; w = 0.0f; }

  int i0 = faces[3 * sel + 0], i1 = faces[3 * sel + 1], i2 = faces[3 * sel + 2];
  float x = u * verts[3 * i0 + 0] + v * verts[3 * i1 + 0] + w * verts[3 * i2 + 0];
  float y = u * verts[3 * i0 + 1] + v * verts[3 * i1 + 1] + w * verts[3 * i2 + 1];
  float z = u * verts[3 * i0 + 2] + v * verts[3 * i1 + 2] + w * verts[3 * i2 + 2];
  float4 o; o.x = x; o.y = y; o.z = z; o.w = x * x + y * y + z * z;
  out[s] = o;
}

// ---------------------------------------------------------------------------
// 4) Min distance via WMMA, fully folded into the matrix op:
//      A row m = (-2x, -2y, -2z, 1),  B col n = (x, y, z, |t|^2)
//      => A·B = |t_n|^2 - 2 s_m·t_n   with C = inline 0 (no accum init).
//    Track min across column tiles; add |s_m|^2 and take one sqrt per row
//    at the end (sqrt/max are monotonic, min commutes with a row constant).
//    8 waves/block, each wave owns one 16-row tile; 2 column tiles per iter
//    (compiler further unrolls + fuses mins into v_min3_num_f32).
// ---------------------------------------------------------------------------
__global__ __launch_bounds__(256)
void psd_min_dist_wmma_kernel(const float4* __restrict__ src,
                              const float4* __restrict__ tgt,
                              float* __restrict__ rowmin, int n) {
  const int lane = threadIdx.x & 31;
  const int wave = threadIdx.x >> 5;
  const int row0 = (blockIdx.x * 8 + wave) * 16;
  const bool hi  = lane >= 16;
  const int  l16 = lane & 15;

  // A 16x4 f32 layout: lanes 0-15 -> (K0,K1)=(-2x,-2y); lanes 16-31 -> (K2,K3)=(-2z,1)
  float4 srow = src[row0 + l16];
  v2f a;
  a.x = hi ? (-2.0f * srow.z) : (-2.0f * srow.x);
  a.y = hi ? 1.0f             : (-2.0f * srow.y);

  // C-tile layout: VGPR r holds row (r + (hi?8:0)); gather those source norms.
  float sn[8];
#pragma unroll
  for (int r = 0; r < 8; ++r) sn[r] = src[row0 + r + (hi ? 8 : 0)].w;

  float rmin[8];  // running min of (|t|^2 - 2 s.t)
#pragma unroll
  for (int r = 0; r < 8; ++r) rmin[r] = 3.0e38f;

  const float4* tp = tgt + l16;
  for (int col0 = 0; col0 < n; col0 += 32) {
    float4 t0 = tp[col0];
    float4 t1 = tp[col0 + 16];
    // B 4x16 f32 layout: lanes 0-15 -> (K0,K1)=(x,y); lanes 16-31 -> (K2,K3)=(z,|t|^2)
    v2f b0, b1;
    b0.x = hi ? t0.z : t0.x;  b0.y = hi ? t0.w : t0.y;
    b1.x = hi ? t1.z : t1.x;  b1.y = hi ? t1.w : t1.y;
    v8f zero = {};  // lowers to inline-0 SRC2
    v8f c0 = __builtin_amdgcn_wmma_f32_16x16x4_f32(false, a, false, b0,
                                                   (short)0, zero, false, false);
    v8f c1 = __builtin_amdgcn_wmma_f32_16x16x4_f32(false, a, false, b1,
                                                   (short)0, zero, false, false);
#pragma unroll
    for (int r = 0; r < 8; ++r)
      rmin[r] = fminf(fminf(rmin[r], c0[r]), c1[r]);  // -> v_min3_num_f32
  }

  // Min over the 16 columns held by each 16-lane half (wave32).
#pragma unroll
  for (int r = 0; r < 8; ++r) {
    float m = rmin[r];
    m = fminf(m, __shfl_xor(m, 1, 16));
    m = fminf(m, __shfl_xor(m, 2, 16));
    m = fminf(m, __shfl_xor(m, 4, 16));
    m = fminf(m, __shfl_xor(m, 8, 16));
    rmin[r] = m;
  }
  if (l16 == 0) {
    int base = row0 + (hi ? 8 : 0);
#pragma unroll
    for (int r = 0; r < 8; ++r)
      rowmin[base + r] = sqrtf(fmaxf(sn[r] + rmin[r], 1e-12f));
  }
}

// ---------------------------------------------------------------------------
// 5) Final scalar: mean(prob)*mean(d_s2o) + mean(d_o2s)
// ---------------------------------------------------------------------------
__global__ __launch_bounds__(256)
void psd_reduce_kernel(const float* __restrict__ d_o2s,
                       const float* __restrict__ d_s2o,
                       const float* __restrict__ prob, int nprob, int nsamp,
                       float* __restrict__ out) {
  __shared__ float sh[256];
  int tid = threadIdx.x;
  float s1 = 0.0f, s2 = 0.0f, s3 = 0.0f;
  for (int i = tid; i < nsamp; i += 256) { s1 += d_o2s[i]; s2 += d_s2o[i]; }
  for (int i = tid; i < nprob; i += 256) s3 += prob[i];
  float sums[3] = {s1, s2, s3};
  float tot[3];
  for (int k = 0; k < 3; ++k) {
    sh[tid] = sums[k];
    __syncthreads();
    for (int off = 128; off > 0; off >>= 1) {
      if (tid < off) sh[tid] += sh[tid + off];
      __syncthreads();
    }
    tot[k] = sh[0];
    __syncthreads();
  }
  if (tid == 0)
    out[0] = (tot[2] / (float)nprob) * (tot[1] / (float)nsamp) +
             tot[0] / (float)nsamp;
}

// ---------------------------------------------------------------------------
extern "C" void kernel_launch(void* const* d_in, const int* in_sizes, int n_in,
                              void* d_out, int out_size, void* d_ws,
                              size_t ws_size, hipStream_t stream) {
  const float* orig_pos   = (const float*)d_in[0];
  const int*   orig_faces = (const int*)d_in[1];
  const float* simp_pos   = (const float*)d_in[2];
  const int*   simp_faces = (const int*)d_in[3];
  const float* prob       = (const float*)d_in[4];
  const float* u_o  = (const float*)d_in[5];
  const float* v_o  = (const float*)d_in[6];
  const float* u_s  = (const float*)d_in[7];
  const float* v_s  = (const float*)d_in[8];
  const float* rs_o = (const float*)d_in[9];
  const float* rs_s = (const float*)d_in[10];

  const int F_o   = in_sizes[1] / 3;
  const int F_s   = in_sizes[3] / 3;
  const int nprob = in_sizes[4];
  const int S     = in_sizes[5];  // 8192
  (void)n_in; (void)out_size; (void)ws_size;

  char* ws = (char*)d_ws;
  size_t off = 0;
  auto alloc = [&](size_t bytes) -> void* {
    void* p = ws + off;
    off = (off + bytes + 255) & ~((size_t)255);
    return p;
  };
  float*  cdf_o  = (float*)alloc((size_t)F_o * sizeof(float));
  float*  cdf_s  = (float*)alloc((size_t)F_s * sizeof(float));
  float4* smp_o  = (float4*)alloc((size_t)S * sizeof(float4));
  float4* smp_s  = (float4*)alloc((size_t)S * sizeof(float4));
  float*  mn_o2s = (float*)alloc((size_t)S * sizeof(float));
  float*  mn_s2o = (float*)alloc((size_t)S * sizeof(float));

  psd_face_area_kernel<<<(F_o + 255) / 256, 256, 0, stream>>>(orig_pos, orig_faces, cdf_o, F_o);
  psd_face_area_kernel<<<(F_s + 255) / 256, 256, 0, stream>>>(simp_pos, simp_faces, cdf_s, F_s);
  psd_scan_kernel<<<1, 1024, 0, stream>>>(cdf_o, F_o);
  psd_scan_kernel<<<1, 1024, 0, stream>>>(cdf_s, F_s);
  psd_sample_kernel<<<(S + 255) / 256, 256, 0, stream>>>(orig_pos, orig_faces, cdf_o, u_o, v_o, rs_o, F_o, S, smp_o);
  psd_sample_kernel<<<(S + 255) / 256, 256, 0, stream>>>(simp_pos, simp_faces, cdf_s, u_s, v_s, rs_s, F_s, S, smp_s);
  // Two passes with swapped operands -> both axis-mins with zero atomics.
  psd_min_dist_wmma_kernel<<<S / 128, 256, 0, stream>>>(smp_o, smp_s, mn_o2s, S);
  psd_min_dist_wmma_kernel<<<S / 128, 256, 0, stream>>>(smp_s, smp_o, mn_s2o, S);
  psd_reduce_kernel<<<1, 256, 0, stream>>>(mn_o2s, mn_s2o, prob, nprob, S, (float*)d_out);
}